// Model_20547123544409
// MI455X (gfx1250) — compile-verified
//
#include <hip/hip_runtime.h>

typedef __attribute__((ext_vector_type(2))) float v2f;
typedef __attribute__((ext_vector_type(8))) float v8f;
typedef __attribute__((ext_vector_type(4))) unsigned int u32x4;
typedef __attribute__((ext_vector_type(4))) int i32x4;
typedef __attribute__((ext_vector_type(8))) int i32x8;

#ifndef __has_builtin
#define __has_builtin(x) 0
#endif
#if __has_builtin(__builtin_amdgcn_tensor_load_to_lds) && __has_builtin(__builtin_amdgcn_s_wait_tensorcnt)
#define HAVE_TDM 1
#else
#define HAVE_TDM 0
#endif

// Problem constants (from reference setup_inputs)
constexpr int Hh   = 24;       // heads
constexpr int Pp   = 64;       // state dim p
constexpr int Nn   = 128;      // state dim n
constexpr int Ss   = 8192;     // sequence
constexpr int Ll   = 64;       // BLOCK_LEN
constexpr int Cc   = Ss / Ll;  // 128 chunks
constexpr int KSEG = 16;       // K-split segments per (b,h)
constexpr int CPS  = Cc / KSEG;// chunks per segment = 8
constexpr int XSTR = 66;       // Xs row stride (floats): even => b64-aligned, conflict-free
constexpr int BSTR = 136;      // Bs row stride (floats): 128 + 8 TDM pad dwords

__global__ void zero_out_kernel(float* __restrict__ out, int n) {
    int i = blockIdx.x * blockDim.x + threadIdx.x;
    if (i < n) out[i] = 0.0f;
}

// Per (b,h): chunk-level suffix sums R[c] = sum_{t >= (c+1)*64} A[b,t,h]
__global__ __launch_bounds__(Cc) void suffix_kernel(const float* __restrict__ A,
                                                    float* __restrict__ R) {
    int bh = blockIdx.x;               // 0 .. b*H-1
    int c  = threadIdx.x;              // 0 .. 127
    __shared__ float cs[Cc];
    const float* Ab = A + (size_t)(bh / Hh) * Ss * Hh + (bh % Hh);
    float s = 0.0f;
    int t0 = c * Ll;
    for (int i = 0; i < Ll; ++i) s += Ab[(size_t)(t0 + i) * Hh];
    cs[c] = s;
    __syncthreads();
    float r = 0.0f;
    for (int cc = c + 1; cc < Cc; ++cc) r += cs[cc];
    R[(size_t)bh * Cc + c] = r;
}

#if HAVE_TDM
// Issue a TDM DMA of one B chunk: 64 rows (t) x 128 floats (n), row stride 3072
// floats in global, into LDS at lds_off with 8-dword padding per 128-dword row
// (=> BSTR=136 floats, breaks LDS bank aliasing of the two K-halves).
__device__ __forceinline__ void tdm_load_B(const float* gsrc, unsigned lds_off) {
    unsigned long long ga = (unsigned long long)(uintptr_t)gsrc;
    u32x4 g0;
    g0[0] = 1u;                                       // count=1, user mode, no gather
    g0[1] = lds_off;                                  // lds_addr (bytes)
    g0[2] = (unsigned)(ga & 0xFFFFFFFFu);             // global_addr[31:0]
    g0[3] = (unsigned)((ga >> 32) & 0x1FFFFFFu)       // global_addr[56:32]
          | (2u << 30);                               // type = 2 ("image")
    i32x8 g1;
    g1[0] = (int)((2u << 16)        // data_size = 4B
          | (1u << 20)              // pad_enable
          | (6u << 22)              // pad_interval: 128 dwords
          | (7u << 25));            // pad_amount: 8 dwords
    g1[1] = (int)((unsigned)Nn << 16);   // tensor_dim0[15:0] = 128
    g1[2] = (int)((unsigned)Ll << 16);   // tensor_dim1[15:0] = 64
    g1[3] = (int)((unsigned)Nn << 16);   // tile_dim0 = 128
    g1[4] = Ll;                          // tile_dim1 = 64
    g1[5] = Hh * Nn;                     // tensor_dim0_stride = 3072 elements
    g1[6] = 0;
    g1[7] = 0;
    i32x4 z4 = {0, 0, 0, 0};             // 2-D tensor: groups 2/3 unused
    i32x8 z8 = {0, 0, 0, 0, 0, 0, 0, 0}; // extra group (clang-23 6-arg form): unused
    __builtin_amdgcn_tensor_load_to_lds(g0, g1, z4, z4, z8, 0);
}
#endif

// Main decay-weighted outer-product GEMM: out[bh,p,n] += sum_t w_t X[t,p] B[t,n]
__global__ __launch_bounds__(256) void chunk_gemm_kernel(const float* __restrict__ X,
                                                         const float* __restrict__ A,
                                                         const float* __restrict__ B,
                                                         const float* __restrict__ R,
                                                         float* __restrict__ out) {
    const int wg   = blockIdx.x;
    const int bh   = wg / KSEG;
    const int seg  = wg % KSEG;
    const int b    = bh / Hh;
    const int h    = bh % Hh;
    const int tid  = threadIdx.x;
    const int wave = tid >> 5;     // 0..7 -> N tile
    const int lane = tid & 31;
    const int half = lane >> 4;    // selects K pair (per ISA 16x4 f32 A/B layout)
    const int l15  = lane & 15;    // M (for A) / N (for B) within tile

    __shared__ float Xs[Pp * XSTR];        // [p][t], weight-scaled
    __shared__ float Bs[2][Ll * BSTR];     // [t][n] double buffer (TDM target)
    __shared__ float wLds[Ll];
    __shared__ float aLds[Ll];

    const float* Xbh = X + (size_t)b * Ss * Hh * Pp + (size_t)h * Pp; // + t*H*P + p
    const float* Bbh = B + (size_t)b * Ss * Hh * Nn + (size_t)h * Nn; // + t*H*N + n
    const float* Abh = A + (size_t)b * Ss * Hh + h;                   // + t*H

    v8f acc0 = {}, acc1 = {}, acc2 = {}, acc3 = {};

    const int c0 = seg * CPS, c1 = c0 + CPS;

#if HAVE_TDM
    // Prologue: DMA the first B chunk while we start staging X.
    if (wave == 0) {
        tdm_load_B(Bbh + (size_t)c0 * Ll * Hh * Nn,
                   (unsigned)(uintptr_t)&Bs[c0 & 1][0]);
    }
#endif

    for (int c = c0; c < c1; ++c) {
        const int t0  = c * Ll;
        const int buf = c & 1;
        // 1) load this chunk's A values
        if (tid < Ll) aLds[tid] = Abh[(size_t)(t0 + tid) * Hh];
        __syncthreads();
        // 2) within-chunk suffix sums -> decay weights w_t = exp(R[c] + suffix)
        if (tid < Ll) {
            float s = 0.0f;
            for (int u = tid + 1; u < Ll; ++u) s += aLds[u];
            wLds[tid] = __expf(R[(size_t)bh * Cc + c] + s);
        }
        __syncthreads();
        // 3) stage weighted X transposed: Xs[p][t] = w_t * X[t,p]
        //    (overlaps with the in-flight TDM DMA of this chunk's B)
        #pragma unroll
        for (int i = 0; i < 16; ++i) {
            int lin = i * 256 + tid;
            int tt = lin >> 6, p = lin & 63;
            float v = Xbh[(size_t)(t0 + tt) * (Hh * Pp) + p];
            Xs[p * XSTR + tt] = v * wLds[tt];
        }
#if HAVE_TDM
        if (wave == 0) __builtin_amdgcn_s_wait_tensorcnt(0);
#else
        // Fallback: manual B staging, same [t][n] layout with BSTR stride
        #pragma unroll
        for (int i = 0; i < 32; ++i) {
            int lin = i * 256 + tid;
            int tt = lin >> 7, n = lin & 127;
            Bs[buf][tt * BSTR + n] = Bbh[(size_t)(t0 + tt) * (Hh * Nn) + n];
        }
#endif
        __syncthreads();
#if HAVE_TDM
        // 4) prefetch next chunk's B into the other buffer; it was last read
        //    in chunk c-1 whose compute finished before this barrier.
        if (wave == 0 && (c + 1) < c1) {
            tdm_load_B(Bbh + (size_t)(c + 1) * Ll * Hh * Nn,
                       (unsigned)(uintptr_t)&Bs[(c + 1) & 1][0]);
        }
#endif
        // 5) fp32 WMMA over K=64 in steps of 4
        const float* __restrict__ Bsb = &Bs[buf][0];
        const int nn = wave * 16 + l15;
        #pragma unroll 4
        for (int k0 = 0; k0 < Ll; k0 += 4) {
            const int kb = k0 + 2 * half;
            v2f bf;
            bf.x = Bsb[(kb + 0) * BSTR + nn];
            bf.y = Bsb[(kb + 1) * BSTR + nn];
            v2f a0 = *(const v2f*)&Xs[( 0 + l15) * XSTR + kb];
            v2f a1 = *(const v2f*)&Xs[(16 + l15) * XSTR + kb];
            v2f a2 = *(const v2f*)&Xs[(32 + l15) * XSTR + kb];
            v2f a3 = *(const v2f*)&Xs[(48 + l15) * XSTR + kb];
            acc0 = __builtin_amdgcn_wmma_f32_16x16x4_f32(false, a0, false, bf, (short)0, acc0, false, false);
            acc1 = __builtin_amdgcn_wmma_f32_16x16x4_f32(false, a1, false, bf, (short)0, acc1, false, false);
            acc2 = __builtin_amdgcn_wmma_f32_16x16x4_f32(false, a2, false, bf, (short)0, acc2, false, false);
            acc3 = __builtin_amdgcn_wmma_f32_16x16x4_f32(false, a3, false, bf, (short)0, acc3, false, false);
        }
        __syncthreads();
    }

    // 6) accumulate split-K partials into out[b,h,p,n] with hardware f32 atomics
    float* obase = out + (size_t)bh * Pp * Nn;
    const int n = wave * 16 + l15;
    #pragma unroll
    for (int r = 0; r < 8; ++r) {
        const int p0 = r + 8 * half;   // C/D layout: reg r, lanes>=16 -> M = r+8
        unsafeAtomicAdd(&obase[(size_t)(p0 +  0) * Nn + n], acc0[r]);
        unsafeAtomicAdd(&obase[(size_t)(p0 + 16) * Nn + n], acc1[r]);
        unsafeAtomicAdd(&obase[(size_t)(p0 + 32) * Nn + n], acc2[r]);
        unsafeAtomicAdd(&obase[(size_t)(p0 + 48) * Nn + n], acc3[r]);
    }
}

extern "C" void kernel_launch(void* const* d_in, const int* in_sizes, int n_in,
                              void* d_out, int out_size, void* d_ws, size_t ws_size,
                              hipStream_t stream) {
    const float* X = (const float*)d_in[0];
    const float* A = (const float*)d_in[1];
    const float* B = (const float*)d_in[2];
    float* out = (float*)d_out;
    float* R   = (float*)d_ws;               // b*H*C floats = 48 KB

    const int b  = in_sizes[0] / (Ss * Hh * Pp);   // batch = 4
    const int BH = b * Hh;                         // 96 (b,h) GEMMs

    zero_out_kernel<<<(out_size + 255) / 256, 256, 0, stream>>>(out, out_size);
    suffix_kernel<<<BH, Cc, 0, stream>>>(A, R);
    chunk_gemm_kernel<<<BH * KSEG, 256, 0, stream>>>(X, A, B, R, out);
}